// ReducedDuplicateLoss_1374389534775
// MI455X (gfx1250) — compile-verified
//
#include <hip/hip_runtime.h>
#include <cstdint>
#include <cstddef>

#define CARD       1027
#define BDIM       8
#define SDIM       8192
#define LROW       (SDIM - 1)        // 8191 positions per batch row
#define NROWS      (BDIM * LROW)     // 65528 rows total
#define IGNORE_IDX (-100)

typedef __attribute__((address_space(1))) int gint_t;
typedef __attribute__((address_space(3))) int lint_t;

// ---------------------------------------------------------------------------
// CDNA5 async global->LDS copy (tracked by ASYNCcnt). Builtin confirmed
// present on this toolchain (round-1 diagnostic came from inside the guard).
// Each call moves 4 B/lane = 128 B/wave.
// ---------------------------------------------------------------------------
__device__ __forceinline__ void async_copy_b32(const float* gsrc, float* ldst) {
#if __has_builtin(__builtin_amdgcn_global_load_async_to_lds_b32)
  __builtin_amdgcn_global_load_async_to_lds_b32(
      (gint_t*)gsrc, (lint_t*)ldst, /*imm offset*/ 0, /*cpol*/ 0);
#else
  unsigned l32 = (unsigned)(unsigned long long)(uintptr_t)ldst; // low 32b = LDS offset
  asm volatile("global_load_async_to_lds_b32 %0, %1, off"
               :: "v"(l32), "v"(gsrc) : "memory");
#endif
}

template <int N>
__device__ __forceinline__ void wait_asynccnt() {
#if __has_builtin(__builtin_amdgcn_s_wait_asynccnt)
  __builtin_amdgcn_s_wait_asynccnt(N);
#else
  asm volatile("s_wait_asynccnt %0" :: "i"(N) : "memory");
#endif
}

// Issue one full row body (1024 floats) as 32 async B32 copies (4 KB).
__device__ __forceinline__ void issue_row(const float* __restrict__ grow,
                                          float* __restrict__ lrow, int lane) {
#pragma unroll
  for (int i = 0; i < 32; ++i)
    async_copy_b32(grow + lane + i * 32, lrow + lane + i * 32);
}

__global__ void __launch_bounds__(128)
rdl_main(const float* __restrict__ logits, const int* __restrict__ labels,
         float* __restrict__ ws, int nWaves, int chunk) {
  __shared__ __align__(16) float sbuf[4][2][1024];   // 32 KB: 4 waves x dbl buf
  const int lane = threadIdx.x & 31;
  const int wv   = threadIdx.x >> 5;
  const int gw   = blockIdx.x * 4 + wv;              // global wave id

  float accL = 0.0f, accM = 0.0f;

  const int r0   = gw * chunk;                       // contiguous rows per wave
  int rend       = r0 + chunk;
  if (rend > NROWS) rend = NROWS;

  if (r0 < rend) {
    // one divide per wave, then purely incremental (b, l, base) bookkeeping
    int b = r0 / LROW;
    int l = r0 - b * LROW;
    const float* gcur = logits + ((size_t)b * SDIM + (size_t)l) * CARD;

    issue_row(gcur, &sbuf[wv][0][0], lane);          // prime buffer 0

    int par = 0;
    for (int r = r0; r < rend; ++r) {
      float* cur = &sbuf[wv][par][0];
      float* nxt = &sbuf[wv][par ^ 1][0];
      const bool wrap = (l == LROW - 1);             // (b,8190) -> (b+1,0)
      const float* gnext = gcur + (wrap ? 2 * CARD : CARD);

      if (r + 1 < rend) {
        issue_row(gnext, nxt, lane);
        wait_asynccnt<32>();   // loads complete in order: current row is done
      } else {
        wait_asynccnt<0>();
      }

      // tail elements 1024..1026 straight from global (rows only 4B aligned)
      const float tail = (lane < 3) ? gcur[1024 + lane] : 0.0f;

      // ---- pass 1: row max (registers hold the row, LDS read once) ----
      float4 v[8];
      float m = (lane < 3) ? tail : -INFINITY;
#pragma unroll
      for (int i = 0; i < 8; ++i) {
        v[i] = *(const float4*)(cur + i * 128 + lane * 4);
        m = fmaxf(m, fmaxf(fmaxf(v[i].x, v[i].y), fmaxf(v[i].z, v[i].w)));
      }
#pragma unroll
      for (int off = 16; off > 0; off >>= 1)
        m = fmaxf(m, __shfl_xor(m, off, 32));

      // ---- pass 2: sum of exp ----
      float s = (lane < 3) ? __expf(tail - m) : 0.0f;
#pragma unroll
      for (int i = 0; i < 8; ++i)
        s += __expf(v[i].x - m) + __expf(v[i].y - m) +
             __expf(v[i].z - m) + __expf(v[i].w - m);
#pragma unroll
      for (int off = 16; off > 0; off >>= 1)
        s += __shfl_xor(s, off, 32);

      // ---- target gather, mask, duplicate-decay weight (wave-uniform) ----
      const int* lab = labels + b * SDIM;
      const int tg = lab[l + 1];
      const int pv = lab[l];
      const float mask = (tg != IGNORE_IDX) ? 1.0f : 0.0f;
      const int   t    = (tg == IGNORE_IDX) ? 0 : tg;
      const float logit_t = (t < 1024) ? cur[t] : gcur[t];
      const float nll = (m + __logf(s)) - logit_t;

      float wgt = 1.0f;
      if (tg == pv && l > 0) {
        // walk back to the last reset; DECAY^7 < MIN_W clips anything deeper
        float w = 0.7f;
        int j = l - 1, cnt = 2;                // invariant: cnt = l - j + 1
        while (j > 0 && cnt < 8 && lab[j + 1] == lab[j]) {
          --j; ++cnt; w *= 0.7f;
        }
        wgt = fmaxf(w, 0.1f);                  // clip at MIN_W
      }

      accL += nll * wgt * mask;
      accM += mask;

      // advance to next contiguous row
      if (wrap) { l = 0; ++b; } else { ++l; }
      gcur = gnext;
      par ^= 1;
    }
  }

  if (lane == 0 && gw < nWaves) {
    ws[2 * gw]     = accL;
    ws[2 * gw + 1] = accM;
  }
}

__global__ void __launch_bounds__(256)
rdl_finalize(const float* __restrict__ ws, float* __restrict__ out, int nWaves) {
  __shared__ float sL[256], sM[256];
  const int t = threadIdx.x;
  float a = 0.0f, b = 0.0f;
  for (int i = t; i < nWaves; i += 256) { a += ws[2 * i]; b += ws[2 * i + 1]; }
  sL[t] = a; sM[t] = b;
  __syncthreads();
#pragma unroll
  for (int stp = 128; stp > 0; stp >>= 1) {
    if (t < stp) { sL[t] += sL[t + stp]; sM[t] += sM[t + stp]; }
    __syncthreads();
  }
  if (t == 0) out[0] = sL[0] / sM[0];
}

extern "C" void kernel_launch(void* const* d_in, const int* in_sizes, int n_in,
                              void* d_out, int out_size, void* d_ws, size_t ws_size,
                              hipStream_t stream) {
  (void)in_sizes; (void)n_in; (void)out_size;
  const float* logits = (const float*)d_in[0];
  const int*   labels = (const int*)d_in[1];   // jax int64 w/o x64 -> int32
  float* out = (float*)d_out;
  float* ws  = (float*)d_ws;

  // size the wave count to the scratch we actually have (2 floats per wave)
  int nWaves = 8192;
  const int maxW = (int)(ws_size / (2 * sizeof(float)));
  if (maxW < nWaves) nWaves = (maxW / 4) * 4;
  if (nWaves < 4) nWaves = 4;
  const int nBlocks = nWaves / 4;              // 128 threads = 4 waves per block
  const int chunk   = (NROWS + nWaves - 1) / nWaves;

  rdl_main<<<nBlocks, 128, 0, stream>>>(logits, labels, ws, nWaves, chunk);
  rdl_finalize<<<1, 256, 0, stream>>>(ws, out, nWaves);
}